// Sampler_8985071583849
// MI455X (gfx1250) — compile-verified
//
#include <hip/hip_runtime.h>
#include <stdint.h>

#define NB      256      // B: rows / blocks
#define NG      65536    // G: elements per row
#define NK      256      // K: top-k
#define EPSF    1e-20f
#define THREADS 256
#define CHUNK   1024
#define NCHUNK  (NG / CHUNK)     // 64
#define PERTH   (NG / THREADS)   // 256
#define CANDCAP 1024
#define NSTAGE  3                // async pipeline depth: 2 chunks in flight

// monotone float <-> sortable-u32 (ascending)
__device__ __forceinline__ uint32_t f2key(float f) {
  uint32_t u = __float_as_uint(f);
  return (u & 0x80000000u) ? ~u : (u | 0x80000000u);
}
__device__ __forceinline__ float key2f(uint32_t k) {
  uint32_t u = (k & 0x80000000u) ? (k ^ 0x80000000u) : ~k;
  return __uint_as_float(u);
}
// generic pointer to LDS: low 32 bits are the workgroup-relative LDS byte offset
__device__ __forceinline__ uint32_t lds_off(const void* p) {
  return (uint32_t)(uintptr_t)p;
}
// CDNA5 async global->LDS copy, ASYNCcnt-tracked; NT hint keeps the streamed
// single-use data from evicting the edges_p table out of the 192MB L2.
__device__ __forceinline__ void async_ld_b32_nt(uint32_t lds_dst, uint64_t gaddr) {
  asm volatile("global_load_async_to_lds_b32 %0, %1, off th:TH_LOAD_NT"
               :: "v"(lds_dst), "v"(gaddr) : "memory");
}

__global__ __launch_bounds__(THREADS) void sampler_topk_kernel(
    const int* __restrict__ inputs,     // [B*G, 2] int32: (eg_idx, edge_id)
    const float* __restrict__ edges_p,  // [8M]
    const float* __restrict__ gumbel,   // [B, G]
    float* __restrict__ out)            // [B*K*2] pairs ++ [B*K] edges_y (as f32)
{
  __shared__ uint32_t keys[NG];                  // 256 KB: whole row resident in LDS
  __shared__ int      stage_e[NSTAGE][CHUNK];    // 12 KB: async-staged edge_id
  __shared__ float    stage_g[NSTAGE][CHUNK];    // 12 KB: async-staged gumbel_u
  __shared__ unsigned long long cand[CANDCAP];   // 8 KB : top-k candidates
  __shared__ uint32_t hist[256];                 // radix histogram
  __shared__ float    redM[THREADS];
  __shared__ float    redS[THREADS];
  __shared__ uint32_t sh_bin, sh_above, sh_cnt;

  const int t   = threadIdx.x;
  const int row = blockIdx.x;
  const uint64_t rowBase = (uint64_t)row * NG;
  const uint64_t in_addr = (uint64_t)(uintptr_t)inputs;
  const uint64_t gu_addr = (uint64_t)(uintptr_t)gumbel;

  auto issue_chunk = [&](int c) {
    const int sb = c % NSTAGE;
#pragma unroll
    for (int q = 0; q < 4; ++q) {
      const int slot = q * THREADS + t;
      const uint64_t e = rowBase + (uint64_t)(c * CHUNK + slot);
      async_ld_b32_nt(lds_off(&stage_e[sb][slot]), in_addr + e * 8u + 4u); // edge_id
      async_ld_b32_nt(lds_off(&stage_g[sb][slot]), gu_addr + e * 4u);      // gumbel
    }
  };

  // ---- Phase A: stream row (2 chunks prefetched), online softmax, keys->LDS ----
  float m = -INFINITY, s_acc = 0.0f;
  issue_chunk(0);
  issue_chunk(1);
  for (int c = 0; c < NCHUNK; ++c) {
    if (c + 2 < NCHUNK) {
      issue_chunk(c + 2);
      asm volatile("s_wait_asynccnt 16" ::: "memory");  // chunk c landed, 2 in flight
    } else if (c + 1 < NCHUNK) {
      asm volatile("s_wait_asynccnt 8" ::: "memory");   // chunk c landed, 1 in flight
    } else {
      asm volatile("s_wait_asynccnt 0" ::: "memory");   // last chunk landed
    }
    const int sb = c % NSTAGE;
#pragma unroll
    for (int q = 0; q < 4; ++q) {
      const int slot = q * THREADS + t;
      const int   eid = stage_e[sb][slot];
      const float u   = stage_g[sb][slot];
      const float p   = edges_p[eid];                    // L2-resident 32MB table
      const float y   = p - logf(-logf(u + EPSF) + EPSF);
      const float nm  = fmaxf(m, y);
      s_acc = s_acc * expf(m - nm) + expf(y - nm);
      m = nm;
      keys[c * CHUNK + slot] = f2key(y);
    }
  }

  // ---- Phase B: reduce (max, sum) across 256 threads ----
  redM[t] = m; redS[t] = s_acc;
  __syncthreads();
  for (int off = THREADS / 2; off > 0; off >>= 1) {
    if (t < off) {
      const float m1 = redM[t], s1 = redS[t];
      const float m2 = redM[t + off], s2 = redS[t + off];
      const float mm = fmaxf(m1, m2);
      redS[t] = s1 * expf(m1 - mm) + s2 * expf(m2 - mm);
      redM[t] = mm;
    }
    __syncthreads();
  }
  const float M = redM[0];
  const float S = redS[0];

  // ---- Phase C: 4-pass MSB radix select over LDS keys (find K-th key T) ----
  uint32_t prefix = 0;
  uint32_t krem = NK;
  for (int shift = 24; shift >= 0; shift -= 8) {
    hist[t] = 0;
    __syncthreads();
    const uint32_t himask = (shift == 24) ? 0u : (0xFFFFFFFFu << (shift + 8));
    for (int j = 0; j < PERTH; ++j) {
      const uint32_t k = keys[j * THREADS + t];
      if ((k & himask) == prefix)
        atomicAdd(&hist[(k >> shift) & 0xFFu], 1u);
    }
    __syncthreads();
    if (t == 0) {
      uint32_t cum = 0; int b = 255;
      for (; b > 0; --b) {
        const uint32_t c = hist[b];
        if (cum + c >= krem) break;
        cum += c;
      }
      sh_bin = (uint32_t)b;
      sh_above = cum;
    }
    __syncthreads();
    prefix |= (sh_bin << shift);
    krem   -= sh_above;
  }
  const uint32_t T = prefix;

  // ---- Phase D: compact all keys >= T, tie-break by lower index ----
  if (t == 0) sh_cnt = 0;
  __syncthreads();
  for (int j = 0; j < PERTH; ++j) {
    const int i = j * THREADS + t;
    const uint32_t k = keys[i];
    if (k >= T) {
      const uint32_t p = atomicAdd(&sh_cnt, 1u);
      if (p < CANDCAP)
        cand[p] = ((unsigned long long)k << 32) |
                  (unsigned long long)(0xFFFFFFFFu - (uint32_t)i);
    }
  }
  __syncthreads();
  const uint32_t ncand = sh_cnt < CANDCAP ? sh_cnt : CANDCAP;
  for (uint32_t p = ncand + t; p < CANDCAP; p += THREADS) cand[p] = 0ull;
  __syncthreads();

  // ---- Phase E: bitonic sort candidates descending (key desc, idx asc) ----
  for (int ksz = 2; ksz <= CANDCAP; ksz <<= 1) {
    for (int jj = ksz >> 1; jj > 0; jj >>= 1) {
      for (int e = t; e < CANDCAP; e += THREADS) {
        const int ixj = e ^ jj;
        if (ixj > e) {
          const unsigned long long a = cand[e], b2 = cand[ixj];
          const bool desc = ((e & ksz) == 0);
          if (desc ? (a < b2) : (a > b2)) { cand[e] = b2; cand[ixj] = a; }
        }
      }
      __syncthreads();
    }
  }

  // ---- Phase F: emit top-K ----
  {
    const unsigned long long c = cand[t];
    const uint32_t k = (uint32_t)(c >> 32);
    const uint32_t i = 0xFFFFFFFFu - (uint32_t)(c & 0xFFFFFFFFull);
    const float y  = key2f(k);
    const float pv = expf(y - M) / S;
    const float ey = (1.0f - pv) + pv;                 // matches reference fp path
    const int eid  = inputs[(size_t)((rowBase + i) << 1) + 1];
    const int o    = row * NK + t;
    out[2 * o + 0] = (float)row;                       // eg_idx[row, :] == row
    out[2 * o + 1] = (float)eid;
    out[2 * NB * NK + o] = ey;
  }
}

extern "C" void kernel_launch(void* const* d_in, const int* in_sizes, int n_in,
                              void* d_out, int out_size, void* d_ws, size_t ws_size,
                              hipStream_t stream) {
  (void)in_sizes; (void)n_in; (void)out_size; (void)d_ws; (void)ws_size;
  const int*   inputs  = (const int*)d_in[0];
  const float* edges_p = (const float*)d_in[1];
  const float* gumbel  = (const float*)d_in[2];
  float*       out     = (float*)d_out;
  sampler_topk_kernel<<<NB, THREADS, 0, stream>>>(inputs, edges_p, gumbel, out);
}